// LIF_47931835023809
// MI455X (gfx1250) — compile-verified
//
#include <hip/hip_runtime.h>
#include <stdint.h>
#include <stddef.h>

// -----------------------------------------------------------------------------
// LIF scan over [B=65536, L=400] f32, thread-per-row, staged through LDS via the
// CDNA5 Tensor Data Mover (tensor_load_to_lds / tensor_store_from_lds).
//
//   chunking: 4 chunks of 100 timesteps, tiles of [256 rows x 100 cols]
//   LDS:      3 buffers x 100KB = 300KB (of 320KB/WGP), triple buffered
//   roles:    wave0 issues TDM loads, wave1 issues TDM stores (per-wave
//             TENSORcnt queues stay independent; barriers hand off buffers)
//   drains:   store queue is drained only before a buffer is reloaded
//             (chunk 0 -> buffer 0 reused by load 3); all remaining stores
//             are covered by S_ENDPGM's implicit S_WAIT_IDLE.
// -----------------------------------------------------------------------------

namespace {
constexpr int kB      = 65536;   // batch rows
constexpr int kL      = 400;     // timesteps
constexpr int kRows   = 256;     // rows per block (== blockDim.x, 8 wave32s)
constexpr int kC      = 100;     // timesteps per chunk
constexpr int kNChunk = 4;       // kL / kC
constexpr int kNBuf   = 3;       // LDS buffers
constexpr int kBufF   = kRows * kC;                 // floats per buffer (25600)
constexpr uint32_t kBufBytes = (uint32_t)kBufF * 4; // 102400 B
}

typedef unsigned int u32x4 __attribute__((ext_vector_type(4)));
typedef int          i32x4 __attribute__((ext_vector_type(4)));
typedef int          i32x8 __attribute__((ext_vector_type(8)));

template <int N>
__device__ __forceinline__ void wait_tensorcnt() {
#if defined(__HIP_DEVICE_COMPILE__)
  __builtin_amdgcn_s_wait_tensorcnt(N);
#endif
}

// Build a D# for a [kRows x kC] f32 tile of the [kB x kL] row-major tensor
// (tensor_dim0 = kL, stride0 = kL, tensor_dim1 = kB) and issue the TDM op.
// global_byte_addr points at the tile start (not the tensor start).
template <bool IS_STORE>
__device__ __forceinline__ void tdm_issue(uint32_t lds_byte_addr,
                                          uint64_t global_byte_addr) {
#if defined(__HIP_DEVICE_COMPILE__)
  // ---- D# group 0 (CDNA5 ISA) ----
  u32x4 g0;
  g0[0] = 1u;                                           // count=1, user desc
  g0[1] = lds_byte_addr;                                // lds_addr
  g0[2] = (uint32_t)(global_byte_addr & 0xFFFFFFFFull); // global_addr[31:0]
  g0[3] = (uint32_t)((global_byte_addr >> 32) & 0x1FFFFFFull)
        | (2u << 30);                                   // type=2 ("image")

  // ---- D# group 1 (§8.4): data_size=4B, dims/strides, tile shape ----
  i32x8 g1;
  g1[0] = (int)(2u << 16);                              // data_size = 4 bytes
  g1[1] = (int)(((uint32_t)kL & 0xFFFFu) << 16);        // tensor_dim0[15:0]
  g1[2] = (int)((((uint32_t)kL >> 16) & 0xFFFFu)        // tensor_dim0[31:16]
        |        (((uint32_t)kB & 0xFFFFu) << 16));     // tensor_dim1[15:0]
  g1[3] = (int)((((uint32_t)kB >> 16) & 0xFFFFu)        // tensor_dim1[31:16]
        |        ((uint32_t)kC << 16));                 // tile_dim0
  g1[4] = (int)((uint32_t)kRows);                       // tile_dim1 (tile_dim2=0)
  g1[5] = (int)((uint32_t)kL);                          // tensor_dim0_stride[31:0]
  g1[6] = 0;                                            // stride0 hi | stride1 lo
  g1[7] = 0;                                            // stride1 hi

  i32x4 z4 = {0, 0, 0, 0};                              // groups 2/3: 2-D tensor

#if __clang_major__ >= 23
  i32x8 z8 = {0, 0, 0, 0, 0, 0, 0, 0};
  if (IS_STORE)
    __builtin_amdgcn_tensor_store_from_lds(g0, g1, z4, z4, z8, 0);
  else
    __builtin_amdgcn_tensor_load_to_lds(g0, g1, z4, z4, z8, 0);
#else
  if (IS_STORE)
    __builtin_amdgcn_tensor_store_from_lds(g0, g1, z4, z4, 0);
  else
    __builtin_amdgcn_tensor_load_to_lds(g0, g1, z4, z4, 0);
#endif
#else
  (void)lds_byte_addr; (void)global_byte_addr;
#endif
}

// One chunk of 100 timesteps. Wave0 waits on its load queue, everyone
// computes in-place (input tile -> spike tile), wave1 stores the tile out.
template <int K>
__device__ __forceinline__ void run_chunk(float* lds, uint32_t lds_base,
                                          uint64_t i_base, uint64_t o_base,
                                          float& v, float& sumS, float& sumTS,
                                          int& first) {
  constexpr int kIssued = (K == 0) ? 3 : 4;   // loads issued before this chunk
  constexpr bool kReuse = (K + 3 < kNChunk);  // stored buffer reloaded later?
  const bool w0 = (threadIdx.x < 32u);
  const bool w1 = (threadIdx.x >= 32u) && (threadIdx.x < 64u);

  if (w0) wait_tensorcnt<kIssued - (K + 1)>();  // loads complete in order
  __syncthreads();                              // tile K ready in LDS

  // --- compute: thread r owns row r; row stride 100 dwords (16B aligned,
  // 2-way bank conflict only), ds_load_b128 / ds_store_b128, in-place spikes.
  float* row = lds + (K % kNBuf) * kBufF + (int)threadIdx.x * kC;
#pragma unroll
  for (int tt = 0; tt < kC; tt += 4) {
    float4 x = *reinterpret_cast<float4*>(row + tt);
    float e[4] = {x.x, x.y, x.z, x.w};
#pragma unroll
    for (int j = 0; j < 4; ++j) {
      const int t = K * kC + tt + j;
      v = __builtin_fmaf(v, 0.95f, e[j]);       // v += -v/TAU + I_t
      const bool sp = (v >= 1.0f);              // s_hard (forward value of s)
      const float s = sp ? 1.0f : 0.0f;
      first = (sp && (first < 0)) ? t : first;  // argmax == first spike
      sumS += s;
      sumTS = __builtin_fmaf(s, (float)t, sumTS);
      v = sp ? 0.0f : v;                        // detached hard reset
      e[j] = s;
    }
    x.x = e[0]; x.y = e[1]; x.z = e[2]; x.w = e[3];
    *reinterpret_cast<float4*>(row + tt) = x;   // spikes staged in place
  }
  __syncthreads();                              // spikes visible, reads done

  if (w1) {
    tdm_issue<true>(lds_base + (uint32_t)(K % kNBuf) * kBufBytes,
                    o_base + 4ull * (uint64_t)(K * kC));
    if constexpr (kReuse) wait_tensorcnt<0>();  // drain only before reuse
  }
  __syncthreads();                              // (store K done, if drained)

  if constexpr (kReuse) {                       // refill the freed buffer
    if (w0) {
      tdm_issue<false>(lds_base + (uint32_t)((K + 3) % kNBuf) * kBufBytes,
                       i_base + 4ull * (uint64_t)((K + 3) * kC));
    }
  }
}

__global__ __launch_bounds__(kRows) void lif_tdm_kernel(
    const float* __restrict__ I, float* __restrict__ out) {
  __shared__ float lds[kNBuf * kBufF];          // 307,200 B of 320 KB

  const uint32_t lds_base  = (uint32_t)(uintptr_t)(&lds[0]);  // LDS byte offset
  const uint64_t first_row = (uint64_t)blockIdx.x * kRows;
  const uint64_t i_base = (uint64_t)(uintptr_t)I   + 4ull * first_row * kL;
  const uint64_t o_base = (uint64_t)(uintptr_t)out + 4ull * first_row * kL;

  if (threadIdx.x < 32u) {                      // wave0: prefetch 3 tiles
    tdm_issue<false>(lds_base + 0u * kBufBytes, i_base);
    tdm_issue<false>(lds_base + 1u * kBufBytes, i_base + 4ull * kC);
    tdm_issue<false>(lds_base + 2u * kBufBytes, i_base + 8ull * kC);
  }

  float v = 0.0f, sumS = 0.0f, sumTS = 0.0f;
  int first = -1;

  run_chunk<0>(lds, lds_base, i_base, o_base, v, sumS, sumTS, first);
  run_chunk<1>(lds, lds_base, i_base, o_base, v, sumS, sumTS, first);
  run_chunk<2>(lds, lds_base, i_base, o_base, v, sumS, sumTS, first);
  run_chunk<3>(lds, lds_base, i_base, o_base, v, sumS, sumTS, first);

  // Tail outputs (coalesced): hard latency then soft latency, after spikes.
  // Remaining tensor stores are drained by S_ENDPGM's implicit S_WAIT_IDLE.
  const size_t b = (size_t)blockIdx.x * kRows + threadIdx.x;
  out[(size_t)kB * kL + b]              = (float)(first < 0 ? 0 : first);
  out[(size_t)kB * kL + (size_t)kB + b] = sumTS / (sumS + 1e-6f);
}

extern "C" void kernel_launch(void* const* d_in, const int* in_sizes, int n_in,
                              void* d_out, int out_size, void* d_ws,
                              size_t ws_size, hipStream_t stream) {
  (void)in_sizes; (void)n_in; (void)out_size; (void)d_ws; (void)ws_size;
  const float* I = (const float*)d_in[0];
  float* out = (float*)d_out;
  lif_tdm_kernel<<<kB / kRows, kRows, 0, stream>>>(I, out);
}